// Mamba2_90769838834224
// MI455X (gfx1250) — compile-verified
//
#include <hip/hip_runtime.h>
#include <hip/hip_bf16.h>

// ---------------------------------------------------------------------------
// Mamba2 block, fused fp32-WMMA pipeline for gfx1250 (MI455X, wave32).
//   K1: xbc = u @ W_in[:, :640]  (WMMA f32 16x16x4, branch-free 5 tiles/wave)
//       + fused dt dot-product -> dA = softplus(dt+bias) * (-exp(A_log))
//   K2: KV[b,h] = B_ssm^T @ (V * dA)   (WMMA, LDS-staged K chunks)
//   K3: y = C @ KV + D*V -> LayerNorm -> out = y @ W_out (all fused, WMMA)
// ---------------------------------------------------------------------------

typedef __attribute__((ext_vector_type(2))) float v2f;
typedef __attribute__((ext_vector_type(8))) float v8f;

#define BATCH   32
#define LSEQ    4096
#define DMODEL  256
#define NPROJ   648          // W_in width: 512 + 64 + 64 + 8
#define XBW     640          // stored width of xbc workspace (x|B|C only)
#define DINNER  512
#define DSTATE  64
#define NHEADS  8
#define HEADDIM 64
#define MTOT    (BATCH * LSEQ)   // 131072

__device__ __forceinline__ v8f wmma4(v2f a, v2f b, v8f c) {
    // V_WMMA_F32_16X16X4_F32: D = A(16x4) * B(4x16) + C(16x16), fp32
    return __builtin_amdgcn_wmma_f32_16x16x4_f32(
        /*neg_a=*/false, a, /*neg_b=*/false, b,
        /*c_mod=*/(short)0, c, /*reuse_a=*/false, /*reuse_b=*/false);
}

// ---------------------------------------------------------------------------
// Kernel 1: in-projection.  grid = MTOT/16, block = 256 (8 waves).
// WMMA part: 40 full col tiles over cols 0..639; wave w owns tiles
// {w, w+8, ..., w+32} -> 5 tiles each, no guards anywhere in the K loop.
// dt part: threads 0..127 compute the 8 dt columns as serial dots and emit
// dA directly (fuses the old elementwise kernel).
// ---------------------------------------------------------------------------
__global__ __launch_bounds__(256) void k_inproj(const float* __restrict__ u,
                                                const float* __restrict__ Win,
                                                const float* __restrict__ dt_bias,
                                                const float* __restrict__ A_log,
                                                float* __restrict__ xb,
                                                float* __restrict__ dAv) {
    const int tid   = threadIdx.x;
    const int lane  = tid & 31;
    const int wave  = tid >> 5;               // 0..7
    const int row16 = lane & 15;
    const int khalf = (lane >> 4) * 2;        // 0 or 2
    const int rbase = (lane >> 4) * 8;
    const size_t mbase = (size_t)blockIdx.x * 16;

    // ---- WMMA main GEMM: 16 x 640 block ----
    v8f acc[5] = {};
    const float* arow = u + (mbase + row16) * DMODEL;
    const int col0 = wave * 16 + row16;       // tile t adds 128*t

    for (int k = 0; k < DMODEL; k += 4) {
        if (k + 32 < DMODEL)
            __builtin_prefetch(arow + k + 32, 0, 0);   // global_prefetch_b8
        v2f a;
        a.x = arow[k + khalf];
        a.y = arow[k + khalf + 1];
        const float* brow0 = Win + (size_t)(k + khalf) * NPROJ + col0;
        const float* brow1 = brow0 + NPROJ;
#pragma unroll
        for (int t = 0; t < 5; ++t) {
            v2f b;
            b.x = brow0[t * 128];
            b.y = brow1[t * 128];
            acc[t] = wmma4(a, b, acc[t]);
        }
    }

#pragma unroll
    for (int t = 0; t < 5; ++t) {
        const int col = col0 + t * 128;
#pragma unroll
        for (int r = 0; r < 8; ++r)
            xb[(mbase + rbase + r) * (size_t)XBW + col] = acc[t][r];
    }

    // ---- fused dt -> dA: 16 rows x 8 heads on threads 0..127 ----
    if (tid < 128) {
        const int r  = tid >> 3;              // 0..15
        const int hh = tid & 7;
        const float* ur = u + (mbase + r) * DMODEL;
        const float* wc = Win + (DINNER + 2 * DSTATE) + hh;   // col 640+hh
        float d0 = 0.f, d1 = 0.f, d2 = 0.f, d3 = 0.f;
#pragma unroll 4
        for (int k = 0; k < DMODEL; k += 4) {
            d0 = fmaf(ur[k + 0], wc[(size_t)(k + 0) * NPROJ], d0);
            d1 = fmaf(ur[k + 1], wc[(size_t)(k + 1) * NPROJ], d1);
            d2 = fmaf(ur[k + 2], wc[(size_t)(k + 2) * NPROJ], d2);
            d3 = fmaf(ur[k + 3], wc[(size_t)(k + 3) * NPROJ], d3);
        }
        const float xv = (d0 + d1) + (d2 + d3) + dt_bias[hh];
        const float sp = (xv > 20.f) ? xv : log1pf(expf(xv));
        dAv[(mbase + r) * NHEADS + hh] = sp * (-expf(A_log[hh]));
    }
}

// ---------------------------------------------------------------------------
// Kernel 2: KV[b,h] (64x64) = sum_l B_ssm[l,s] * (V[l,p] * dA[l,h]).
// grid = BATCH*NHEADS, block = 512 (16 waves = 4x4 tiles of 16x16).
// 64-row K chunks staged in LDS (padded stride 65).
// ---------------------------------------------------------------------------
#define KCHUNK 64
__global__ __launch_bounds__(512) void k_kv(const float* __restrict__ xb,
                                            const float* __restrict__ dAv,
                                            float* __restrict__ KV) {
    __shared__ float Bs[KCHUNK][65];
    __shared__ float Vs[KCHUNK][65];

    const int b    = blockIdx.x >> 3;
    const int h    = blockIdx.x & 7;
    const int tid  = threadIdx.x;
    const int lane = tid & 31;
    const int wave = tid >> 5;            // 0..15
    const int st   = wave >> 2;           // s tile 0..3
    const int pt   = wave & 3;            // p tile 0..3
    const int row16 = lane & 15;
    const int khalf = (lane >> 4) * 2;
    const size_t bL = (size_t)b * LSEQ;

    v8f acc = {};

    for (int l0 = 0; l0 < LSEQ; l0 += KCHUNK) {
        __syncthreads();
#pragma unroll
        for (int j = 0; j < (KCHUNK * 64) / 512; ++j) {
            const int idx = tid + j * 512;
            const int ll  = idx >> 6;
            const int cc  = idx & 63;
            const size_t roff = (bL + l0 + ll) * (size_t)XBW;
            Bs[ll][cc] = xb[roff + DINNER + cc];
            const float sc = dAv[(bL + l0 + ll) * NHEADS + h];
            Vs[ll][cc] = xb[roff + h * HEADDIM + cc] * sc;
        }
        __syncthreads();
#pragma unroll 4
        for (int kk = 0; kk < KCHUNK; kk += 4) {
            v2f a, bb;
            a.x  = Bs[kk + khalf][st * 16 + row16];
            a.y  = Bs[kk + khalf + 1][st * 16 + row16];
            bb.x = Vs[kk + khalf][pt * 16 + row16];
            bb.y = Vs[kk + khalf + 1][pt * 16 + row16];
            acc = wmma4(a, bb, acc);
        }
    }

    const int rbase = (lane >> 4) * 8;
    float* dst = KV + (size_t)(b * NHEADS + h) * DSTATE * HEADDIM;
#pragma unroll
    for (int r = 0; r < 8; ++r) {
        const int srow = st * 16 + rbase + r;
        const int pcol = pt * 16 + row16;
        dst[srow * HEADDIM + pcol] = acc[r];
    }
}

// ---------------------------------------------------------------------------
// Kernel 3: fused tail.  grid = MTOT/16, block = 256 (wave = head).
//   y(16x512) = C(16x64) @ KV[b,h](64x64) + D[h]*V   -> LDS
//   LayerNorm per row (wave32 shfl_xor reduction)
//   out(16x256) = y_ln(16x512) @ W_out(512x256)
// ---------------------------------------------------------------------------
#define SPITCH 516   // 512 + 4: row stride spreads rows across LDS banks
__global__ __launch_bounds__(256) void k_tail(const float* __restrict__ xb,
                                              const float* __restrict__ KV,
                                              const float* __restrict__ Dp,
                                              const float* __restrict__ gamma,
                                              const float* __restrict__ beta,
                                              const float* __restrict__ Wout,
                                              float* __restrict__ out) {
    __shared__ float ysm[16 * SPITCH];

    const int mblk = blockIdx.x;
    const int b    = mblk >> 8;                 // 256 blocks per batch
    const size_t mbase = (size_t)mblk * 16;
    const int tid  = threadIdx.x;
    const int lane = tid & 31;
    const int h    = tid >> 5;                  // wave = head 0..7
    const int row16 = lane & 15;
    const int khalf = (lane >> 4) * 2;
    const int rbase = (lane >> 4) * 8;

    // ---- Phase A: y = C @ KV[b,h] over this head's 64 columns ----
    v8f acc[4] = {};
    const float* crow = xb + (mbase + row16) * (size_t)XBW + (DINNER + DSTATE);
    const float* kvb  = KV + (size_t)(b * NHEADS + h) * DSTATE * HEADDIM;

    for (int k = 0; k < DSTATE; k += 4) {
        v2f a;
        a.x = crow[k + khalf];
        a.y = crow[k + khalf + 1];
#pragma unroll
        for (int t = 0; t < 4; ++t) {
            v2f bb;
            bb.x = kvb[(k + khalf) * HEADDIM + t * 16 + row16];
            bb.y = kvb[(k + khalf + 1) * HEADDIM + t * 16 + row16];
            acc[t] = wmma4(a, bb, acc[t]);
        }
    }

    // ---- + D*V, spill to LDS ----
    const float Dh = Dp[h];
#pragma unroll
    for (int t = 0; t < 4; ++t) {
        const int col = h * HEADDIM + t * 16 + row16;   // 0..511
#pragma unroll
        for (int r = 0; r < 8; ++r) {
            const float v = acc[t][r] +
                Dh * xb[(mbase + rbase + r) * (size_t)XBW + col];
            ysm[(rbase + r) * SPITCH + col] = v;
        }
    }
    __syncthreads();

    // ---- Phase C: LayerNorm; wave h handles rows 2h and 2h+1 ----
#pragma unroll
    for (int q = 0; q < 2; ++q) {
        const int rr = 2 * h + q;
        float s1 = 0.f, s2 = 0.f;
#pragma unroll
        for (int j = 0; j < 16; ++j) {
            const float v = ysm[rr * SPITCH + lane + 32 * j];
            s1 += v; s2 += v * v;
        }
#pragma unroll
        for (int m = 16; m >= 1; m >>= 1) {
            s1 += __shfl_xor(s1, m, 32);
            s2 += __shfl_xor(s2, m, 32);
        }
        const float mu   = s1 * (1.f / DINNER);
        const float var  = s2 * (1.f / DINNER) - mu * mu;
        const float rstd = rsqrtf(var + 1e-5f);
#pragma unroll
        for (int j = 0; j < 16; ++j) {
            const int col = lane + 32 * j;
            const float v = ysm[rr * SPITCH + col];
            ysm[rr * SPITCH + col] = (v - mu) * rstd * gamma[col] + beta[col];
        }
    }
    __syncthreads();

    // ---- Phase D: out = y_ln @ W_out; wave h owns col tiles h and h+8 ----
    v8f oacc[2] = {};
    for (int k = 0; k < DINNER; k += 4) {
        v2f a;
        a.x = ysm[row16 * SPITCH + k + khalf];
        a.y = ysm[row16 * SPITCH + k + khalf + 1];
        const float* w0 = Wout + (size_t)(k + khalf) * DMODEL;
        const float* w1 = w0 + DMODEL;
#pragma unroll
        for (int t = 0; t < 2; ++t) {
            const int col = (h + 8 * t) * 16 + row16;
            v2f bb;
            bb.x = w0[col];
            bb.y = w1[col];
            oacc[t] = wmma4(a, bb, oacc[t]);
        }
    }
#pragma unroll
    for (int t = 0; t < 2; ++t) {
        const int col = (h + 8 * t) * 16 + row16;
#pragma unroll
        for (int r = 0; r < 8; ++r)
            out[(mbase + rbase + r) * DMODEL + col] = oacc[t][r];
    }
}

// ---------------------------------------------------------------------------
extern "C" void kernel_launch(void* const* d_in, const int* in_sizes, int n_in,
                              void* d_out, int out_size, void* d_ws, size_t ws_size,
                              hipStream_t stream) {
    const float* x       = (const float*)d_in[0];
    const float* W_in    = (const float*)d_in[1];
    const float* dt_bias = (const float*)d_in[2];
    const float* A_log   = (const float*)d_in[3];
    const float* D_param = (const float*)d_in[4];
    const float* ln_g    = (const float*)d_in[5];
    const float* ln_b    = (const float*)d_in[6];
    const float* W_out   = (const float*)d_in[7];
    float* out = (float*)d_out;

    // Workspace layout (floats): xbc [MTOT x 640], dA [MTOT x 8],
    // KV [BATCH x NHEADS x 64 x 64]  => ~328 MiB total.
    float* xb  = (float*)d_ws;
    float* dAv = xb  + (size_t)MTOT * XBW;
    float* KVw = dAv + (size_t)MTOT * NHEADS;

    k_inproj<<<MTOT / 16, 256, 0, stream>>>(x, W_in, dt_bias, A_log, xb, dAv);
    k_kv<<<BATCH * NHEADS, 512, 0, stream>>>(xb, dAv, KVw);
    k_tail<<<MTOT / 16, 256, 0, stream>>>(xb, KVw, D_param, ln_g, ln_b, W_out, out);
}